// Attention_20315195310831
// MI455X (gfx1250) — compile-verified
//
#include <hip/hip_runtime.h>
#include <hip/hip_bf16.h>

// ---------------------------------------------------------------------------
// CDNA5 (gfx1250) fp32 attention block using V_WMMA_F32_16X16X4_F32.
// wave32 layouts per CDNA5 ISA 7.12.2:
//   A (16x4 f32): lanes 0-15 -> M=lane, VGPR i -> K = 2*half + i
//   B (4x16 f32): lanes -> N=lane%16,   VGPR i -> K = 2*half + i
//   C/D (16x16):  VGPR r -> M = r + 8*half, N = lane%16
// ---------------------------------------------------------------------------

typedef __attribute__((ext_vector_type(2))) float v2f;
typedef __attribute__((ext_vector_type(8))) float v8f;

#define B_ 2
#define T_ 2048
#define C_ 1024
#define H_ 16
#define D_ 64
#define HD_ (H_ * D_)
#define EPS_ 1e-5f
#define SCALE_ 0.125f  // 1/sqrt(64)

// Async global->LDS copy (CDNA5 GLOBAL_LOAD_ASYNC_TO_LDS_B128) via inline asm:
// the clang builtin types its pointer args with language address spaces that
// plain HIP pointers cannot convert to, so emit the instruction directly.
// VGLOBAL async form: vdst = per-lane LDS byte address, vaddr = 64b address.
#if defined(__gfx1250__)
#define USE_ASYNC_LDS 1
#endif

static __device__ __forceinline__ void async_copy_b128(const float* gsrc,
                                                       void* ldst) {
#if defined(USE_ASYNC_LDS)
  const unsigned lds_addr = (unsigned)(unsigned long long)ldst;  // addr[31:0]
  asm volatile("global_load_async_to_lds_b128 %0, %1, off"
               :
               : "v"(lds_addr), "v"(gsrc)
               : "memory");
#else
  *(float4*)ldst = *(const float4*)gsrc;
#endif
}

static __device__ __forceinline__ void async_wait() {
#if defined(USE_ASYNC_LDS)
  asm volatile("s_wait_asynccnt 0x0" ::: "memory");
#endif
}

static __device__ __forceinline__ v8f wmma_f32(v2f a, v2f b, v8f c) {
  // 8 args: (neg_a, A, neg_b, B, c_mod, C, reuse_a, reuse_b)
  return __builtin_amdgcn_wmma_f32_16x16x4_f32(false, a, false, b, (short)0, c,
                                               false, false);
}

static __device__ __forceinline__ v8f v8zero() {
  v8f z = {0.f, 0.f, 0.f, 0.f, 0.f, 0.f, 0.f, 0.f};
  return z;
}

// ---------------------------------------------------------------------------
// Tiled fp32 GEMM: C[M,N] = A[M,K] * W[K,N].  Block tile 128x64, Kc = 32,
// double-buffered LDS software pipeline:
//   iter i: issue async-A(i+1) + global-B(i+1) loads -> compute(i) ->
//           scatter B(i+1) regs to LDS -> s_wait_asynccnt + barrier.
// 8 waves; wave w owns a 32x32 patch (2x2 WMMA subtiles -> 4 WMMAs per
// {a0,a1,b0,b1} fragment quad).  B tile is pair-interleaved in LDS so each
// B-frag is one ds_load_b64 into an even-aligned pair; B row stride is padded
// to 160 floats (== 32 mod 64 banks) so the two half-waves hit disjoint banks.
// Requires M%128==0, N%64==0, K%32==0 (true for all uses here).
// ---------------------------------------------------------------------------
__global__ __launch_bounds__(256) void gemm_xw(const float* __restrict__ A,
                                               const float* __restrict__ W,
                                               float* __restrict__ Cout,
                                               int N, int K) {
  __shared__ float As[2][128][36];   // 36-float row stride: conflict-free
  __shared__ float2 Bs[2][16][80];   // [k/2][n] = {B[k][n], B[k+1][n]}

  const int tid = threadIdx.x;
  const int lane = tid & 31;
  const int w = tid >> 5;
  const int half = lane >> 4;
  const int l = lane & 15;

  const int m0 = blockIdx.y * 128;
  const int n0 = blockIdx.x * 64;
  const int mw = (w & 3) * 32;   // wave's 32-row band
  const int nw = (w >> 2) * 32;  // wave's 32-col band

  // per-thread cooperative-load coordinates
  const int ra = tid >> 3, ca = (tid & 7) * 4;   // A: rows ra + 32*qd
  const int rb = tid >> 4, cb = (tid & 15) * 4;  // B: rows rb + 16*qd

  v8f c00 = v8zero(), c01 = v8zero(), c10 = v8zero(), c11 = v8zero();
  float4 breg[2];

  // ---- prologue: stage tile 0
#pragma unroll
  for (int qd = 0; qd < 4; ++qd)
    async_copy_b128(&A[(size_t)(m0 + ra + qd * 32) * K + ca],
                    &As[0][ra + qd * 32][ca]);
#pragma unroll
  for (int qd = 0; qd < 2; ++qd)
    breg[qd] = *(const float4*)&W[(size_t)(rb + qd * 16) * N + n0 + cb];
#pragma unroll
  for (int qd = 0; qd < 2; ++qd) {
    const int r = rb + qd * 16;
    float* bsrow = (float*)&Bs[0][r >> 1][0] + (r & 1);
    bsrow[(cb + 0) * 2] = breg[qd].x;
    bsrow[(cb + 1) * 2] = breg[qd].y;
    bsrow[(cb + 2) * 2] = breg[qd].z;
    bsrow[(cb + 3) * 2] = breg[qd].w;
  }
  async_wait();
  __syncthreads();

  const int ntiles = K >> 5;
  for (int it = 0; it < ntiles; ++it) {
    const int buf = it & 1;
    const bool have_next = (it + 1 < ntiles);

    // ---- stage tile it+1 (A: async direct to LDS; B: into VGPRs for now)
    if (have_next) {
      const int k0 = (it + 1) << 5;
#pragma unroll
      for (int qd = 0; qd < 4; ++qd)
        async_copy_b128(&A[(size_t)(m0 + ra + qd * 32) * K + k0 + ca],
                        &As[buf ^ 1][ra + qd * 32][ca]);
#pragma unroll
      for (int qd = 0; qd < 2; ++qd)
        breg[qd] =
            *(const float4*)&W[(size_t)(k0 + rb + qd * 16) * N + n0 + cb];
    }

    // ---- compute tile it: 8 kk-steps x 4 WMMAs
#pragma unroll
    for (int kk = 0; kk < 32; kk += 4) {
      const v2f a0 = *(const v2f*)&As[buf][mw + l][kk + 2 * half];
      const v2f a1 = *(const v2f*)&As[buf][mw + 16 + l][kk + 2 * half];
      const v2f b0 = *(const v2f*)&Bs[buf][(kk >> 1) + half][nw + l];
      const v2f b1 = *(const v2f*)&Bs[buf][(kk >> 1) + half][nw + 16 + l];
      c00 = wmma_f32(a0, b0, c00);
      c01 = wmma_f32(a0, b1, c01);
      c10 = wmma_f32(a1, b0, c10);
      c11 = wmma_f32(a1, b1, c11);
    }

    if (have_next) {
      // ---- scatter staged B regs into the other buffer, pair-interleaved
#pragma unroll
      for (int qd = 0; qd < 2; ++qd) {
        const int r = rb + qd * 16;
        float* bsrow = (float*)&Bs[buf ^ 1][r >> 1][0] + (r & 1);
        bsrow[(cb + 0) * 2] = breg[qd].x;
        bsrow[(cb + 1) * 2] = breg[qd].y;
        bsrow[(cb + 2) * 2] = breg[qd].z;
        bsrow[(cb + 3) * 2] = breg[qd].w;
      }
      async_wait();     // tile it+1 A-copies landed
      __syncthreads();  // tile it+1 B-stores visible; tile it reads done
    }
  }

#pragma unroll
  for (int r = 0; r < 8; ++r) {
    const int row0 = m0 + mw + r + 8 * half;
    const int row1 = row0 + 16;
    Cout[(size_t)row0 * N + n0 + nw + l] = c00[r];
    Cout[(size_t)row0 * N + n0 + nw + 16 + l] = c01[r];
    Cout[(size_t)row1 * N + n0 + nw + l] = c10[r];
    Cout[(size_t)row1 * N + n0 + nw + 16 + l] = c11[r];
  }
}

// ---------------------------------------------------------------------------
// Per-head RMSNorm over D=64, in place.  One wave per (b,t,h) group.
// Memory is [B*T, H*D] so group g occupies 64 contiguous floats at g*64.
// ---------------------------------------------------------------------------
__global__ __launch_bounds__(256) void rmsnorm_heads(float* __restrict__ p,
                                                     const float* __restrict__ w,
                                                     int ngroups) {
  const int g = blockIdx.x * 8 + (threadIdx.x >> 5);
  if (g >= ngroups) return;  // whole waves only (ngroups % 8 == 0)
  const int lane = threadIdx.x & 31;
  float2 xv = *(float2*)&p[(size_t)g * 64 + lane * 2];
  float ss = xv.x * xv.x + xv.y * xv.y;
#pragma unroll
  for (int m = 16; m >= 1; m >>= 1) ss += __shfl_xor(ss, m, 32);
  const float r = rsqrtf(ss * (1.0f / 64.0f) + EPS_);
  const float2 wv = *(const float2*)&w[lane * 2];
  xv.x *= r * wv.x;
  xv.y *= r * wv.y;
  *(float2*)&p[(size_t)g * 64 + lane * 2] = xv;
}

// ---------------------------------------------------------------------------
// Causal flash attention, one wave per (b, h, 16-query block).
//   S = (Q*SCALE) K^T via 16 WMMAs (K-depth 64), online softmax,
//   P -> LDS -> A-frags, O += P V via 16 WMMAs (4 d-tiles x 4 k-steps).
// Epilogue: y = gate * O / l  + v  (value residual), layout [B*T, H*D].
// ---------------------------------------------------------------------------
__global__ __launch_bounds__(32) void flash_attn(const float* __restrict__ q,
                                                 const float* __restrict__ k,
                                                 const float* __restrict__ v,
                                                 const float* __restrict__ gate,
                                                 float* __restrict__ y) {
  __shared__ float pl[16 * 16];

  const int lane = threadIdx.x & 31;
  const int half = lane >> 4;
  const int l = lane & 15;
  const int t0 = blockIdx.x * 16;
  const int h = blockIdx.y;
  const int b = blockIdx.z;

  // Q fragments (row = t0 + l, pre-scaled by 1/sqrt(D))
  const size_t rowQ = ((size_t)(b * T_ + t0 + l)) * HD_ + h * D_;
  v2f aQ[16];
#pragma unroll
  for (int t = 0; t < 16; ++t) {
    const float2 qv = *(const float2*)&q[rowQ + t * 4 + 2 * half];
    aQ[t].x = qv.x * SCALE_;
    aQ[t].y = qv.y * SCALE_;
  }

  float mrow[8], lrow[8];
  v8f o[4];
#pragma unroll
  for (int r = 0; r < 8; ++r) {
    mrow[r] = -__builtin_inff();
    lrow[r] = 0.f;
  }
#pragma unroll
  for (int t = 0; t < 4; ++t) o[t] = v8zero();

  for (int j0 = 0; j0 <= t0; j0 += 16) {
    // ---- S tile = Q K^T : B-frag b[i] = K[j0+l][kk + 2*half + i]
    const size_t rowK = ((size_t)(b * T_ + j0 + l)) * HD_ + h * D_;
    v8f s = v8zero();
#pragma unroll
    for (int t = 0; t < 16; ++t) {
      const float2 kv = *(const float2*)&k[rowK + t * 4 + 2 * half];
      v2f bK;
      bK.x = kv.x;
      bK.y = kv.y;
      s = wmma_f32(aQ[t], bK, s);
    }

    // ---- causal mask on the diagonal block (key j0+l vs query t0+m)
    if (j0 == t0) {
#pragma unroll
      for (int r = 0; r < 8; ++r) {
        const int m = r + 8 * half;
        if (l > m) s[r] = -__builtin_inff();
      }
    }

    // ---- online softmax (rows live across the 16 lanes of each half)
    float p[8], f[8];
#pragma unroll
    for (int r = 0; r < 8; ++r) {
      float rm = s[r];
#pragma unroll
      for (int msk = 8; msk >= 1; msk >>= 1)
        rm = fmaxf(rm, __shfl_xor(rm, msk, 32));
      const float mnew = fmaxf(mrow[r], rm);
      p[r] = expf(s[r] - mnew);
      float rs = p[r];
#pragma unroll
      for (int msk = 8; msk >= 1; msk >>= 1) rs += __shfl_xor(rs, msk, 32);
      f[r] = expf(mrow[r] - mnew);
      lrow[r] = lrow[r] * f[r] + rs;
      mrow[r] = mnew;
    }
#pragma unroll
    for (int t = 0; t < 4; ++t)
#pragma unroll
      for (int r = 0; r < 8; ++r) o[t][r] *= f[r];

    // ---- C-layout P -> LDS -> A-layout frags
    __syncthreads();  // protect pl from previous iteration's readers
#pragma unroll
    for (int r = 0; r < 8; ++r) pl[(r + 8 * half) * 16 + l] = p[r];
    __syncthreads();
    v2f aP[4];
#pragma unroll
    for (int t = 0; t < 4; ++t) {
      aP[t].x = pl[l * 16 + t * 4 + 2 * half];
      aP[t].y = pl[l * 16 + t * 4 + 2 * half + 1];
    }

    // ---- O += P V : B-frag b[i] = V[j0 + kq*4 + 2*half + i][d0 + l]
#pragma unroll
    for (int dt = 0; dt < 4; ++dt) {
#pragma unroll
      for (int kq = 0; kq < 4; ++kq) {
        const size_t rowV =
            ((size_t)(b * T_ + j0 + kq * 4 + 2 * half)) * HD_ + h * D_ +
            dt * 16 + l;
        v2f bV;
        bV.x = v[rowV];
        bV.y = v[rowV + HD_];
        o[dt] = wmma_f32(aP[kq], bV, o[dt]);
      }
    }
  }

  // ---- epilogue: y = gate * O / l + v
  const float g = gate[h];
#pragma unroll
  for (int r = 0; r < 8; ++r) {
    const float inv = g / lrow[r];
    const int m = r + 8 * half;
    const size_t rowO = ((size_t)(b * T_ + t0 + m)) * HD_ + h * D_;
#pragma unroll
    for (int dt = 0; dt < 4; ++dt) {
      const size_t idx = rowO + dt * 16 + l;
      y[idx] = o[dt][r] * inv + v[idx];
    }
  }
}

// ---------------------------------------------------------------------------
extern "C" void kernel_launch(void* const* d_in, const int* in_sizes, int n_in,
                              void* d_out, int out_size, void* d_ws,
                              size_t ws_size, hipStream_t stream) {
  (void)in_sizes; (void)n_in; (void)out_size;
  const float* x    = (const float*)d_in[0];
  // d_in[1] = attention_mask (causal; applied analytically)
  const float* Wq   = (const float*)d_in[2];
  const float* Wk   = (const float*)d_in[3];
  const float* Wv   = (const float*)d_in[4];
  const float* Wo   = (const float*)d_in[5];
  const float* qw   = (const float*)d_in[6];
  const float* kw   = (const float*)d_in[7];
  const float* gate = (const float*)d_in[8];
  float* out = (float*)d_out;

  const size_t nBT = (size_t)B_ * T_;        // 4096 rows
  const size_t sz  = nBT * (size_t)HD_;      // 4M floats per buffer
  if (ws_size < 4 * sz * sizeof(float)) return;  // needs 64 MB scratch
  float* qb = (float*)d_ws;
  float* kb = qb + sz;
  float* vb = kb + sz;
  float* yb = vb + sz;

  const dim3 gg(C_ / 64, (unsigned)(nBT / 128));  // 16 x 32 blocks
  gemm_xw<<<gg, 256, 0, stream>>>(x, Wq, qb, HD_, C_);
  gemm_xw<<<gg, 256, 0, stream>>>(x, Wk, kb, HD_, C_);
  gemm_xw<<<gg, 256, 0, stream>>>(x, Wv, vb, HD_, C_);

  const int ngroups = (int)(nBT * H_);  // 65536
  rmsnorm_heads<<<ngroups / 8, 256, 0, stream>>>(qb, qw, ngroups);
  rmsnorm_heads<<<ngroups / 8, 256, 0, stream>>>(kb, kw, ngroups);

  const dim3 ga(T_ / 16, H_, B_);  // 128 x 16 x 2 waves
  flash_attn<<<ga, 32, 0, stream>>>(qb, kb, vb, gate, yb);

  gemm_xw<<<gg, 256, 0, stream>>>(yb, Wo, out, C_, C_);
}